// BaseGraphNeuralNetworkLayer_27298812133763
// MI455X (gfx1250) — compile-verified
//
#include <hip/hip_runtime.h>
#include <math.h>

// Problem constants (match reference)
#define N_NODES   10000
#define KNBR      32
#define DATOM     128
#define DDIR      64
#define DDIRIN    10
#define DDIST     128
#define DDISTIN   128

typedef __attribute__((ext_vector_type(2))) float v2f;
typedef __attribute__((ext_vector_type(8))) float v8f;

__device__ __forceinline__ float silu_f(float x) {
    return x * (1.0f / (1.0f + __expf(-x)));
}

// Kernel 1: sd = silu(nde @ srcW + srcB), td = silu(nde @ tgtW + tgtB)
// N*64 threads, each computes one (n, j) for both matrices (reuses nde loads).
__global__ __launch_bounds__(256)
void dir_embed_kernel(const float* __restrict__ nde,
                      const float* __restrict__ sW, const float* __restrict__ sB,
                      const float* __restrict__ tW, const float* __restrict__ tB,
                      float* __restrict__ sd, float* __restrict__ td) {
    int t = blockIdx.x * blockDim.x + threadIdx.x;
    if (t >= N_NODES * DDIR) return;
    int n = t / DDIR;
    int j = t - n * DDIR;
    float as = sB[j];
    float at = tB[j];
#pragma unroll
    for (int i = 0; i < DDIRIN; ++i) {
        float x = nde[n * DDIRIN + i];
        as += x * sW[i * DDIR + j];
        at += x * tW[i * DDIR + j];
    }
    sd[t] = silu_f(as);
    td[t] = silu_f(at);
}

// Kernel 2: one block (256 thr = 8 waves) per node.
//  - edist = silu(E[32x128] @ distW[128x128] + b) via V_WMMA_F32_16X16X4_F32,
//    masked row-sum fused in registers -> out cols [0,128)
//  - gathered sd / src_atom sums       -> out cols [128,320)
//  - broadcast td / tgt_atom           -> out cols [320,512)
__global__ __launch_bounds__(256)
void edge_agg_kernel(const int* __restrict__ an,
                     const float* __restrict__ ede,
                     const int* __restrict__ nbrs,
                     const unsigned char* __restrict__ nmask,
                     const float* __restrict__ src_atom,
                     const float* __restrict__ tgt_atom,
                     const float* __restrict__ distW,
                     const float* __restrict__ distB,
                     const float* __restrict__ sd,
                     const float* __restrict__ td,
                     float* __restrict__ out) {
    __shared__ float sE[KNBR][132];   // 132 = 128 + 4 pad -> conflict-free A reads
    __shared__ float sM[KNBR];
    __shared__ int   sN[KNBR];
    __shared__ int   sAN[KNBR];

    const int n   = blockIdx.x;
    const int tid = threadIdx.x;

    // --- stage neighbor metadata + E tile into LDS (coalesced) ---
    if (tid < KNBR) {
        int nb   = nbrs[n * KNBR + tid];
        sN[tid]  = nb;
        sM[tid]  = nmask[n * KNBR + tid] ? 1.0f : 0.0f;
        sAN[tid] = an[nb];
    }
    const float* erow = ede + (size_t)n * (KNBR * DDISTIN);
    for (int idx = tid; idx < KNBR * DDISTIN; idx += 256) {
        sE[idx >> 7][idx & 127] = erow[idx];
    }
    __syncthreads();

    float sumM = 0.0f;
#pragma unroll
    for (int k = 0; k < KNBR; ++k) sumM += sM[k];
    const float inv = 1.0f / (sumM + 1e-5f);

    // --- WMMA part: cols [0,128) ---
    const int wave    = tid >> 5;
    const int lane    = tid & 31;
    const int half    = lane >> 4;   // 0: lanes 0-15, 1: lanes 16-31
    const int lrow    = lane & 15;
    const int colbase = wave * 16;
    const int col     = colbase + lrow;

    v8f acc0 = {};   // edge rows 0..15
    v8f acc1 = {};   // edge rows 16..31

    for (int kk = 0; kk < DDISTIN; kk += 4) {
        const int k0 = kk + 2 * half;   // per-lane K pair per ISA 16x4 layout
        v2f a0, a1, b;
        a0.x = sE[lrow][k0];
        a0.y = sE[lrow][k0 + 1];
        a1.x = sE[16 + lrow][k0];
        a1.y = sE[16 + lrow][k0 + 1];
        b.x  = distW[k0 * DDIST + col];
        b.y  = distW[(k0 + 1) * DDIST + col];
        acc0 = __builtin_amdgcn_wmma_f32_16x16x4_f32(
                   false, a0, false, b, (short)0, acc0, false, false);
        acc1 = __builtin_amdgcn_wmma_f32_16x16x4_f32(
                   false, a1, false, b, (short)0, acc1, false, false);
    }

    // Epilogue: bias + silu + masked row reduction (D layout: lane holds
    // column `col`, rows v + 8*half in VGPR v).
    const float bias = distB[col];
    float csum = 0.0f;
#pragma unroll
    for (int v = 0; v < 8; ++v) {
        const int r = v + 8 * half;
        csum += sM[r]      * silu_f(acc0[v] + bias);
        csum += sM[16 + r] * silu_f(acc1[v] + bias);
    }
    csum += __shfl_xor(csum, 16, 32);   // merge the two half-tiles (same col)
    if (lane < 16) {
        out[(size_t)n * 512 + col] = csum * inv;
    }

    // --- gather + broadcast parts: cols [128,512) ---
    const int an_n = sAN[0] * 0 + an[n];   // scalar broadcast load
    const size_t obase = (size_t)n * 512;
    for (int c = tid; c < 384; c += 256) {
        if (c < 64) {                       // sender_dir -> cols 128..191
            float s = 0.0f;
#pragma unroll 4
            for (int k = 0; k < KNBR; ++k)
                s += sM[k] * sd[(size_t)sN[k] * DDIR + c];
            out[obase + 128 + c] = s * inv;
        } else if (c < 192) {               // sender_atom -> cols 192..319
            const int cc = c - 64;
            float s = 0.0f;
#pragma unroll 4
            for (int k = 0; k < KNBR; ++k)
                s += sM[k] * src_atom[sAN[k] * DATOM + cc];
            out[obase + 192 + cc] = s * inv;
        } else if (c < 256) {               // recv_dir -> cols 320..383
            const int cc = c - 192;
            out[obase + 320 + cc] = td[(size_t)n * DDIR + cc] * sumM * inv;
        } else {                            // recv_atom -> cols 384..511
            const int cc = c - 256;
            out[obase + 384 + cc] = tgt_atom[an_n * DATOM + cc] * sumM * inv;
        }
    }
}

extern "C" void kernel_launch(void* const* d_in, const int* in_sizes, int n_in,
                              void* d_out, int out_size, void* d_ws, size_t ws_size,
                              hipStream_t stream) {
    const int*           an  = (const int*)d_in[0];
    const float*         nde = (const float*)d_in[1];
    const float*         ede = (const float*)d_in[2];
    const int*           nbr = (const int*)d_in[3];
    const unsigned char* nm  = (const unsigned char*)d_in[4];  // jnp bool = 1 byte
    const float*         sae = (const float*)d_in[5];
    const float*         tae = (const float*)d_in[6];
    const float*         sW  = (const float*)d_in[7];
    const float*         sB  = (const float*)d_in[8];
    const float*         tW  = (const float*)d_in[9];
    const float*         tB  = (const float*)d_in[10];
    const float*         dW  = (const float*)d_in[11];
    const float*         dB  = (const float*)d_in[12];

    float* out = (float*)d_out;
    float* sd  = (float*)d_ws;                       // [N, 64]
    float* td  = sd + (size_t)N_NODES * DDIR;        // [N, 64]  (5.12 MB total)

    dir_embed_kernel<<<(N_NODES * DDIR + 255) / 256, 256, 0, stream>>>(
        nde, sW, sB, tW, tB, sd, td);

    edge_agg_kernel<<<N_NODES, 256, 0, stream>>>(
        an, ede, nbr, nm, sae, tae, dW, dB, sd, td, out);
}